// RecursiveNN_46351287058649
// MI455X (gfx1250) — compile-verified
//
#include <hip/hip_runtime.h>
#include <math.h>

// ---------------------------------------------------------------------------
// RecursiveNN on gfx1250 — f32 WMMA (V_WMMA_F32_16X16X4_F32) tree reduction.
// ---------------------------------------------------------------------------

typedef __attribute__((ext_vector_type(2))) float v2f;
typedef __attribute__((ext_vector_type(8))) float v8f;

#define EMBED 256
#define KDIM  512        // 2*EMBED
#define KC    128        // K chunk staged in LDS
#define MT    64         // block M tile
#define LDST  132        // LDS row stride in floats (128 + 4 pad -> conflict free)

// ---------------------------------------------------------------------------
// Wave-level GEMM core: one wave computes a 32(M) x 64(N) tile.
// A fragment from LDS (staged chunk), B fragment directly from W_w (L2-resident).
// acc[s][t]: s = M subtile (0..1, 16 rows each), t = N subtile (0..3, 16 cols).
// ---------------------------------------------------------------------------
__device__ __forceinline__ void wave_mma_kc(const float* __restrict__ Ww,
                                            const float* smem,
                                            int wm, int wn, int kglob,
                                            v8f acc[2][4])
{
    const int lane = threadIdx.x & 31;
    const int lhi  = lane >> 4;   // 0..1  -> selects K pair (K0/K1 vs K2/K3)
    const int llo  = lane & 15;   // row / col within 16

    for (int kk = 0; kk < KC; kk += 4) {
        const int kl = kk + 2 * lhi;
        v2f a0 = *(const v2f*)&smem[(wm * 32 +      llo) * LDST + kl];
        v2f a1 = *(const v2f*)&smem[(wm * 32 + 16 + llo) * LDST + kl];
        const int kg = kglob + kl;
        v2f b0 = *(const v2f*)&Ww[(wn * 64 +  0 + llo) * KDIM + kg];
        v2f b1 = *(const v2f*)&Ww[(wn * 64 + 16 + llo) * KDIM + kg];
        v2f b2 = *(const v2f*)&Ww[(wn * 64 + 32 + llo) * KDIM + kg];
        v2f b3 = *(const v2f*)&Ww[(wn * 64 + 48 + llo) * KDIM + kg];

        acc[0][0] = __builtin_amdgcn_wmma_f32_16x16x4_f32(false, a0, false, b0, (short)0, acc[0][0], false, false);
        acc[0][1] = __builtin_amdgcn_wmma_f32_16x16x4_f32(false, a0, false, b1, (short)0, acc[0][1], false, false);
        acc[0][2] = __builtin_amdgcn_wmma_f32_16x16x4_f32(false, a0, false, b2, (short)0, acc[0][2], false, false);
        acc[0][3] = __builtin_amdgcn_wmma_f32_16x16x4_f32(false, a0, false, b3, (short)0, acc[0][3], false, false);
        acc[1][0] = __builtin_amdgcn_wmma_f32_16x16x4_f32(false, a1, false, b0, (short)0, acc[1][0], false, false);
        acc[1][1] = __builtin_amdgcn_wmma_f32_16x16x4_f32(false, a1, false, b1, (short)0, acc[1][1], false, false);
        acc[1][2] = __builtin_amdgcn_wmma_f32_16x16x4_f32(false, a1, false, b2, (short)0, acc[1][2], false, false);
        acc[1][3] = __builtin_amdgcn_wmma_f32_16x16x4_f32(false, a1, false, b3, (short)0, acc[1][3], false, false);
    }
}

// D tile layout (ISA 7.12.2): VGPR r -> M = r (lanes 0-15) / r+8 (lanes 16-31),
// N = lane&15 within the 16-wide subtile.  Epilogue: +bias, ReLU, store.
__device__ __forceinline__ void wave_store(float* __restrict__ out,
                                           const float* __restrict__ Wb,
                                           int mbase, int wm, int wn,
                                           v8f acc[2][4])
{
    const int lane = threadIdx.x & 31;
    const int lhi  = lane >> 4;
    const int llo  = lane & 15;
#pragma unroll
    for (int s = 0; s < 2; ++s) {
#pragma unroll
        for (int t = 0; t < 4; ++t) {
            const int n    = wn * 64 + t * 16 + llo;
            const float bv = Wb[n];
#pragma unroll
            for (int r = 0; r < 8; ++r) {
                const int m = mbase + wm * 32 + s * 16 + r + 8 * lhi;
                float v = acc[s][t][r] + bv;
                out[m * EMBED + n] = v > 0.f ? v : 0.f;
            }
        }
    }
}

// ---------------------------------------------------------------------------
// Generic level kernel: A (Mrows x 512 row-major, = previous level flattened,
// pair-concat is free since x[b,2n,:]||x[b,2n+1,:] is contiguous).
// Block: 256 threads = 8 waves in 2(M) x 4(N) arrangement. Tile 64 x 256.
// ---------------------------------------------------------------------------
__global__ __launch_bounds__(256) void level_gemm_kernel(
    const float* __restrict__ A, const float* __restrict__ Ww,
    const float* __restrict__ Wb, float* __restrict__ out)
{
    __shared__ float smem[MT * LDST];
    const int mbase = blockIdx.x * MT;
    const int wave  = threadIdx.x >> 5;
    const int wm = wave >> 2, wn = wave & 3;

    v8f acc[2][4];
#pragma unroll
    for (int s = 0; s < 2; ++s)
#pragma unroll
        for (int t = 0; t < 4; ++t)
            acc[s][t] = (v8f){0.f,0.f,0.f,0.f,0.f,0.f,0.f,0.f};

    for (int c = 0; c < KDIM / KC; ++c) {
        // cooperative stage: 64 rows x 128 cols of A -> LDS (8 float4 / thread)
        {
            const int row  = threadIdx.x >> 2;
            const int col0 = (threadIdx.x & 3) * 32;
            const float4* src = (const float4*)&A[(size_t)(mbase + row) * KDIM + c * KC + col0];
            float4* dst = (float4*)&smem[row * LDST + col0];
#pragma unroll
            for (int i = 0; i < 8; ++i) dst[i] = src[i];
        }
        __syncthreads();
        wave_mma_kc(Ww, smem, wm, wn, c * KC, acc);
        __syncthreads();
    }
    wave_store(out, Wb, mbase, wm, wn, acc);
}

// ---------------------------------------------------------------------------
// Fused embedding-gather + level-1 kernel: A rows are relu(emb[token]) pairs,
// gathered straight into LDS (skips the 268MB x0 round trip through HBM).
// Node row r -> b = r>>9, n = r&511; chunk c: token = tok[b,2n + (c>>1)],
// element base = (c&1)*128.
// ---------------------------------------------------------------------------
__global__ __launch_bounds__(256) void embed_level1_kernel(
    const int* __restrict__ tok, const float* __restrict__ emb,
    const float* __restrict__ Ww, const float* __restrict__ Wb,
    float* __restrict__ out)
{
    __shared__ float smem[MT * LDST];
    const int mbase = blockIdx.x * MT;
    const int wave  = threadIdx.x >> 5;
    const int wm = wave >> 2, wn = wave & 3;

    v8f acc[2][4];
#pragma unroll
    for (int s = 0; s < 2; ++s)
#pragma unroll
        for (int t = 0; t < 4; ++t)
            acc[s][t] = (v8f){0.f,0.f,0.f,0.f,0.f,0.f,0.f,0.f};

    for (int c = 0; c < KDIM / KC; ++c) {
        {
            const int row  = threadIdx.x >> 2;
            const int col0 = (threadIdx.x & 3) * 32;
            const int r = mbase + row;
            const int b = r >> 9;          // 512 nodes per batch row at level 1
            const int n = r & 511;
            const int t = tok[b * 1024 + 2 * n + (c >> 1)];
            const float4* src = (const float4*)&emb[(size_t)t * EMBED + (c & 1) * KC + col0];
            float4* dst = (float4*)&smem[row * LDST + col0];
#pragma unroll
            for (int i = 0; i < 8; ++i) {
                float4 v = src[i];
                v.x = v.x > 0.f ? v.x : 0.f;
                v.y = v.y > 0.f ? v.y : 0.f;
                v.z = v.z > 0.f ? v.z : 0.f;
                v.w = v.w > 0.f ? v.w : 0.f;
                dst[i] = v;
            }
        }
        __syncthreads();
        wave_mma_kc(Ww, smem, wm, wn, c * KC, acc);
        __syncthreads();
    }
    wave_store(out, Wb, mbase, wm, wn, acc);
}

// ---------------------------------------------------------------------------
// Classifier head: logits = root @ P_w^T + P_b; argmax; -log_softmax[label].
// d_out = [prediction(256) as float | loss(256)].
// ---------------------------------------------------------------------------
__global__ __launch_bounds__(256) void head_kernel(
    const float* __restrict__ root, const int* __restrict__ labels,
    const float* __restrict__ Pw, const float* __restrict__ Pb,
    float* __restrict__ out)
{
    const int b = blockIdx.x * blockDim.x + threadIdx.x;
    if (b >= 256) return;
    const float* x = root + (size_t)b * EMBED;
    float l0 = Pb[0], l1 = Pb[1];
    for (int e = 0; e < EMBED; ++e) {
        const float v = x[e];
        l0 += v * Pw[e];
        l1 += v * Pw[EMBED + e];
    }
    const int pred = (l1 > l0) ? 1 : 0;   // argmax, first index wins ties
    const float m  = fmaxf(l0, l1);
    const float lse = m + logf(expf(l0 - m) + expf(l1 - m));
    const int lab = labels[b];
    out[b]       = (float)pred;
    out[256 + b] = lse - (lab == 0 ? l0 : l1);
}

// ---------------------------------------------------------------------------
extern "C" void kernel_launch(void* const* d_in, const int* in_sizes, int n_in,
                              void* d_out, int out_size, void* d_ws, size_t ws_size,
                              hipStream_t stream)
{
    const int*   tok    = (const int*)d_in[0];
    const int*   labels = (const int*)d_in[1];
    const float* emb    = (const float*)d_in[2];
    const float* Ww     = (const float*)d_in[3];
    const float* Wb     = (const float*)d_in[4];
    const float* Pw     = (const float*)d_in[5];
    const float* Pb     = (const float*)d_in[6];
    float* out = (float*)d_out;

    // ping-pong activation buffers: x1 = 256*512*256 floats (134MB),
    // x2.. fit in 67MB second region.
    float* bufA = (float*)d_ws;
    float* bufB = bufA + (size_t)256 * 512 * 256;   // 33,554,432 floats

    // level 1 (fused gather): M = 256 * 512 = 131072 node rows
    embed_level1_kernel<<<131072 / MT, 256, 0, stream>>>(tok, emb, Ww, Wb, bufA);

    // levels 2..10: M = 65536, 32768, ..., 256
    float* cur = bufA;
    float* nxt = bufB;
    int Mout = 65536;
    for (int lvl = 2; lvl <= 10; ++lvl) {
        level_gemm_kernel<<<Mout / MT, 256, 0, stream>>>(cur, Ww, Wb, nxt);
        float* t = cur; cur = nxt; nxt = t;
        Mout >>= 1;
    }

    // head: `cur` holds the 256 x 256 root matrix
    head_kernel<<<1, 256, 0, stream>>>(cur, labels, Pw, Pb, out);
}